// DualCrossAttentionLoss_6605659701627
// MI455X (gfx1250) — compile-verified
//
#include <hip/hip_runtime.h>
#include <math.h>

typedef __attribute__((ext_vector_type(2))) float v2f;
typedef __attribute__((ext_vector_type(8))) float v8f;

#define BB 1024
#define CC 32000
#define DD 2048
#define EPS_SM 0.1f
#define MARGIN_V 0.3f

#define LSTRIDE 36          // 32 K-chunk floats + 4 pad: 144B rows, conflict-free & 16B aligned
#define AS_FLOATS (128 * LSTRIDE)
#define BS_FLOATS (64 * LSTRIDE)
#define BUF_FLOATS (AS_FLOATS + BS_FLOATS)   // 6912 floats per buffer

// workspace layout:
//  ws[0..2]            : CE accumulators (sa, glca, pwca)
//  ws[3..7]            : spare
//  (u32) ws[8..1031]   : encoded hardest-positive keys per anchor (atomicMax)
//  (u32) ws[1032..2055]: encoded hardest-negative keys per anchor (atomicMin)
//  ws[2056..3079]      : squared row norms

__device__ __forceinline__ unsigned fkey(float f) {
    unsigned b = __float_as_uint(f);
    return (b & 0x80000000u) ? ~b : (b | 0x80000000u);   // monotonic float->uint
}
__device__ __forceinline__ float funkey(unsigned k) {
    unsigned b = (k & 0x80000000u) ? (k & 0x7fffffffu) : ~k;
    return __uint_as_float(b);
}

__global__ __launch_bounds__(1024) void init_kernel(float* __restrict__ acc,
                                                    unsigned* __restrict__ wsPos,
                                                    unsigned* __restrict__ wsNeg) {
    const int t = threadIdx.x;
    if (t < 8) acc[t] = 0.0f;
    wsPos[t] = 0x007FFFFFu;   // fkey(-inf)
    wsNeg[t] = 0xFF800000u;   // fkey(+inf)
}

__global__ __launch_bounds__(256) void rownorm_kernel(const float* __restrict__ F,
                                                      float* __restrict__ sq) {
    const int row = blockIdx.x;
    const float4* Fr = (const float4*)(F + (size_t)row * DD);
    float acc = 0.0f;
    for (int j = threadIdx.x; j < DD / 4; j += 256) {
        float4 v = Fr[j];
        acc += v.x * v.x + v.y * v.y + v.z * v.z + v.w * v.w;
    }
    __shared__ float s[256];
    s[threadIdx.x] = acc;
    __syncthreads();
    for (int o = 128; o > 0; o >>= 1) {
        if (threadIdx.x < o) s[threadIdx.x] += s[threadIdx.x + o];
        __syncthreads();
    }
    if (threadIdx.x == 0) sq[row] = s[0];
}

// LDS-tiled Gram GEMM, fused with hardest-pos/neg mining.
// Grid: 8 x 16 blocks of 128(M) x 64(N). Block: 256 threads = 8 waves.
// Wave (wr = w>>1 in 0..3, wc = w&1 in 0..1) owns a 32x32 register block (2x2 WMMA tiles).
// Double-buffered async-to-LDS staging: issue chunk t+1, s_wait_asynccnt 6 keeps
// only the newest group in flight (async loads complete in order).
__global__ __launch_bounds__(256) void triplet_kernel(const float* __restrict__ F,
                                                      const int* __restrict__ labels,
                                                      const float* __restrict__ sq,
                                                      unsigned* __restrict__ wsPos,
                                                      unsigned* __restrict__ wsNeg) {
    __shared__ __align__(16) float smem[2 * BUF_FLOATS];

    const int tid  = threadIdx.x;
    const int w    = tid >> 5;
    const int lane = tid & 31;
    const int half = lane >> 4;   // K-pair select within fragment
    const int lm   = lane & 15;   // lane-in-16-group (row for A frag, col for B frag)
    const int wr   = w >> 1;      // 0..3 : which 32-row strip
    const int wc   = w & 1;       // 0..1 : which 32-col strip

    const int bi  = blockIdx.x >> 4;    // 0..7   (128-row block)
    const int bj  = blockIdx.x & 15;    // 0..15  (64-col block)
    const int m0  = bi * 128;
    const int n0c = bj * 64;

    // staging decomposition (same every chunk)
    const int srowA = tid >> 3;         // +u*32 rows, 4 iterations
    const int sc4   = (tid & 7) * 4;

    auto stage = [&](int bufbase, int kc) {
#pragma unroll
        for (int u = 0; u < 4; ++u) {   // A panel: 128 rows x 32 floats
            const int row = srowA + u * 32;
            const float* g = F + (size_t)(m0 + row) * DD + kc + sc4;
            const unsigned la = (unsigned)(uintptr_t)&smem[bufbase + row * LSTRIDE + sc4];
            asm volatile("global_load_async_to_lds_b128 %0, %1, off"
                         :: "v"(la), "v"((unsigned long long)(uintptr_t)g) : "memory");
        }
#pragma unroll
        for (int u = 0; u < 2; ++u) {   // B panel: 64 rows x 32 floats
            const int row = srowA + u * 32;
            const float* g = F + (size_t)(n0c + row) * DD + kc + sc4;
            const unsigned la = (unsigned)(uintptr_t)&smem[bufbase + AS_FLOATS + row * LSTRIDE + sc4];
            asm volatile("global_load_async_to_lds_b128 %0, %1, off"
                         :: "v"(la), "v"((unsigned long long)(uintptr_t)g) : "memory");
        }
    };

    v8f c[2][2];
#pragma unroll
    for (int i = 0; i < 2; ++i)
#pragma unroll
        for (int j = 0; j < 2; ++j) c[i][j] = (v8f){0.f,0.f,0.f,0.f,0.f,0.f,0.f,0.f};

    stage(0, 0);
    for (int kc = 0; kc < DD; kc += 32) {
        const int bufbase = ((kc >> 5) & 1) ? BUF_FLOATS : 0;
        if (kc + 32 < DD) {
            stage(bufbase ^ BUF_FLOATS, kc + 32);               // prefetch next chunk
            asm volatile("s_wait_asynccnt 6" ::: "memory");     // older 6 (this chunk) done
        } else {
            asm volatile("s_wait_asynccnt 0" ::: "memory");
        }
        __syncthreads();

        const float* As = smem + bufbase;
        const float* Bs = smem + bufbase + AS_FLOATS;
#pragma unroll
        for (int kk = 0; kk < 32; kk += 4) {
            v2f a0 = *(const v2f*)&As[(wr * 32 +  0 + lm) * LSTRIDE + kk + 2 * half];
            v2f a1 = *(const v2f*)&As[(wr * 32 + 16 + lm) * LSTRIDE + kk + 2 * half];
            v2f b0 = *(const v2f*)&Bs[(wc * 32 +  0 + lm) * LSTRIDE + kk + 2 * half];
            v2f b1 = *(const v2f*)&Bs[(wc * 32 + 16 + lm) * LSTRIDE + kk + 2 * half];
            c[0][0] = __builtin_amdgcn_wmma_f32_16x16x4_f32(false, a0, false, b0, (short)0, c[0][0], false, false);
            c[0][1] = __builtin_amdgcn_wmma_f32_16x16x4_f32(false, a0, false, b1, (short)0, c[0][1], false, false);
            c[1][0] = __builtin_amdgcn_wmma_f32_16x16x4_f32(false, a1, false, b0, (short)0, c[1][0], false, false);
            c[1][1] = __builtin_amdgcn_wmma_f32_16x16x4_f32(false, a1, false, b1, (short)0, c[1][1], false, false);
        }
        __syncthreads();   // also protects the buffer re-staged in the next iteration
    }

    // ---- distances + hardest-pos/neg mining in registers ----
    float posm[2][8], negm[2][8];
#pragma unroll
    for (int i = 0; i < 2; ++i)
#pragma unroll
        for (int v = 0; v < 8; ++v) { posm[i][v] = -INFINITY; negm[i][v] = INFINITY; }

    int nlab[2]; float nsq[2]; int nidx[2];
#pragma unroll
    for (int j = 0; j < 2; ++j) {
        nidx[j] = n0c + wc * 32 + j * 16 + lm;
        nlab[j] = labels[nidx[j]];
        nsq[j]  = sq[nidx[j]];
    }
#pragma unroll
    for (int i = 0; i < 2; ++i) {
#pragma unroll
        for (int v = 0; v < 8; ++v) {
            const int   m  = m0 + wr * 32 + i * 16 + 8 * half + v;
            const int   ml = labels[m];
            const float ms = sq[m];
#pragma unroll
            for (int j = 0; j < 2; ++j) {
                const float d2 = ms + nsq[j] - 2.0f * c[i][j][v];
                const float d  = sqrtf(fmaxf(d2, 1e-12f));
                const bool  eq = (ml == nlab[j]);
                if (eq && (m != nidx[j])) posm[i][v] = fmaxf(posm[i][v], d);
                if (!eq)                  negm[i][v] = fminf(negm[i][v], d);
            }
        }
    }

    // reduce over the 16 lanes sharing each row (lane index = column N)
#pragma unroll
    for (int i = 0; i < 2; ++i)
#pragma unroll
        for (int v = 0; v < 8; ++v)
            for (int off = 8; off > 0; off >>= 1) {
                posm[i][v] = fmaxf(posm[i][v], __shfl_xor(posm[i][v], off, 16));
                negm[i][v] = fminf(negm[i][v], __shfl_xor(negm[i][v], off, 16));
            }

    float* redp = smem;         // reuse LDS (all frag reads done; barrier below)
    float* redn = smem + 256;
    __syncthreads();
    if (lm == 0) {
#pragma unroll
        for (int i = 0; i < 2; ++i)
#pragma unroll
            for (int v = 0; v < 8; ++v) {
                const int slot = i * 16 + 8 * half + v;   // 0..31 within wave strip
                redp[w * 32 + slot] = posm[i][v];
                redn[w * 32 + slot] = negm[i][v];
            }
    }
    __syncthreads();
    if (tid < 128) {
        const int wr2 = tid >> 5;
        const int off = tid & 31;
        const float p  = fmaxf(redp[(wr2 * 2 + 0) * 32 + off], redp[(wr2 * 2 + 1) * 32 + off]);
        const float ng = fminf(redn[(wr2 * 2 + 0) * 32 + off], redn[(wr2 * 2 + 1) * 32 + off]);
        atomicMax(&wsPos[m0 + tid], fkey(p));
        atomicMin(&wsNeg[m0 + tid], fkey(ng));
    }
}

// One block per (row, tensor). Single pass; ~1 transcendental per element.
__global__ __launch_bounds__(256) void ce_kernel(const float* __restrict__ sa,
                                                 const float* __restrict__ glca,
                                                 const float* __restrict__ pwca,
                                                 const int* __restrict__ labels,
                                                 float* __restrict__ acc) {
    const int row   = blockIdx.x;
    const int which = blockIdx.y;
    const float* base = (which == 0) ? sa : ((which == 1) ? glca : pwca);
    const float4* xr  = (const float4*)(base + (size_t)row * CC);
    const int tgt = labels[row];

    float m = -INFINITY, s = 0.0f, sumx = 0.0f, xt = 0.0f;
    for (int j = threadIdx.x; j < CC / 4; j += 256) {
        float4 v = xr[j];
        float e[4] = {v.x, v.y, v.z, v.w};
#pragma unroll
        for (int k = 0; k < 4; ++k) {
            const float x = e[k];
            sumx += x;
            if (4 * j + k == tgt) xt = x;
            if (x > m) { s *= __expf(m - x); m = x; }   // rare rescale
            s += __expf(x - m);
        }
    }

    __shared__ float sm[256], ss[256], sx[256], st[256];
    const int tid = threadIdx.x;
    sm[tid] = m; ss[tid] = s; sx[tid] = sumx; st[tid] = xt;
    __syncthreads();
    for (int o = 128; o > 0; o >>= 1) {
        if (tid < o) {
            const float m2 = sm[tid + o], s2 = ss[tid + o];
            const float mn = fmaxf(sm[tid], m2);
            ss[tid] = ss[tid] * expf(sm[tid] - mn) + s2 * expf(m2 - mn);
            sm[tid] = mn;
            sx[tid] += sx[tid + o];
            st[tid] += st[tid + o];
        }
        __syncthreads();
    }
    if (tid == 0) {
        const float lse = sm[0] + logf(ss[0]);
        const float off = EPS_SM / (float)(CC - 1);
        const float contrib = -off * (sx[0] - (float)CC * lse)
                              - (1.0f - EPS_SM - off) * (st[0] - lse);
        atomicAdd(&acc[which], contrib);
    }
}

__global__ __launch_bounds__(1024) void final_kernel(const float* __restrict__ log_vars,
                                                     const float* __restrict__ acc,
                                                     const unsigned* __restrict__ wsPos,
                                                     const unsigned* __restrict__ wsNeg,
                                                     float* __restrict__ out) {
    const int t = threadIdx.x;
    const float p  = funkey(wsPos[t]);
    const float ng = funkey(wsNeg[t]);
    const bool valid = (p > -INFINITY) && (ng < INFINITY);
    float per = fmaxf(p - ng + MARGIN_V, 0.0f);
    if (!valid) per = 0.0f;

    __shared__ float r0[1024], r1[1024], r2[1024];
    r0[t] = per;
    r1[t] = valid ? 1.0f : 0.0f;
    r2[t] = (valid && per > 0.0f) ? 1.0f : 0.0f;
    __syncthreads();
    for (int o = 512; o > 0; o >>= 1) {
        if (t < o) { r0[t] += r0[t + o]; r1[t] += r1[t + o]; r2[t] += r2[t + o]; }
        __syncthreads();
    }
    if (t == 0) {
        const float trip = r0[0] / fmaxf(r1[0], 1.0f);
        const float sa   = acc[0] / (float)BB + trip;
        const float glca = acc[1] / (float)BB;
        const float pwca = acc[2] / (float)BB;
        float lv[3];
        for (int i = 0; i < 3; ++i) lv[i] = fminf(fmaxf(log_vars[i], -5.0f), 5.0f);
        const float total = 0.5f * (expf(-lv[0]) * sa   + lv[0] +
                                    expf(-lv[1]) * glca + lv[1] +
                                    expf(-lv[2]) * pwca + lv[2]);
        out[0] = total;
        out[1] = sa;
        out[2] = glca;
        out[3] = pwca;
        out[4] = r2[0];   // num_active
    }
}

extern "C" void kernel_launch(void* const* d_in, const int* in_sizes, int n_in,
                              void* d_out, int out_size, void* d_ws, size_t ws_size,
                              hipStream_t stream) {
    const float* sa     = (const float*)d_in[0];
    const float* glca   = (const float*)d_in[1];
    const float* pwca   = (const float*)d_in[2];
    const float* feat   = (const float*)d_in[3];
    const int*   labels = (const int*)d_in[4];
    const float* logv   = (const float*)d_in[5];

    float*    ws    = (float*)d_ws;
    float*    acc   = ws;                              // 8 floats
    unsigned* wsPos = (unsigned*)(ws + 8);             // 1024 keys
    unsigned* wsNeg = (unsigned*)(ws + 8 + 1024);      // 1024 keys
    float*    sq    = ws + 8 + 2048;                   // 1024 floats

    init_kernel<<<1, 1024, 0, stream>>>(acc, wsPos, wsNeg);
    rownorm_kernel<<<BB, 256, 0, stream>>>(feat, sq);
    triplet_kernel<<<128, 256, 0, stream>>>(feat, labels, sq, wsPos, wsNeg);
    dim3 grid_ce(BB, 3);
    ce_kernel<<<grid_ce, 256, 0, stream>>>(sa, glca, pwca, labels, acc);
    final_kernel<<<1, 1024, 0, stream>>>(logv, acc, wsPos, wsNeg, (float*)d_out);
}